// Decoder_21354577395995
// MI455X (gfx1250) — compile-verified
//
#include <hip/hip_runtime.h>
#include <hip/hip_bf16.h>
#include <math.h>

// Problem constants (match reference)
constexpr int V_  = 30000;
constexpr int E_  = 300;
constexpr int H_  = 512;
constexpr int B_  = 32;
constexpr int T_  = 32;
constexpr int S_  = 400;
constexpr int OOV_ = 50;
constexpr int VE_ = V_ + OOV_;       // 30050
constexpr float NEG_INF = -1e12f;

typedef float v2f __attribute__((ext_vector_type(2)));
typedef float v8f __attribute__((ext_vector_type(8)));

__device__ __forceinline__ float sigmoidf_(float x) {
    return 1.0f / (1.0f + expf(-x));
}

// ---------------------------------------------------------------------------
// init: h = h0, c = c0, ctx = 0
// ---------------------------------------------------------------------------
__global__ void init_state(const float* __restrict__ h0, const float* __restrict__ c0,
                           float* __restrict__ h, float* __restrict__ c,
                           float* __restrict__ ctx) {
    int i = blockIdx.x * blockDim.x + threadIdx.x;
    if (i < B_ * H_) {
        h[i] = h0[i];
        c[i] = c0[i];
        ctx[i] = 0.0f;
    }
}

// ---------------------------------------------------------------------------
// memories[B*S, H] = encoder_outputs[B*S, H] @ W_enc[H, H] + b_enc
// fp32 WMMA 16x16x4; each wave computes a 32x32 super-tile (2x2 fragments)
// so every A/B fragment load feeds two WMMAs (halves L2 traffic).
// ---------------------------------------------------------------------------
__global__ void memories_wmma(const float* __restrict__ A,     // [12800, 512]
                              const float* __restrict__ Wb,    // [512, 512]
                              const float* __restrict__ bias,  // [512]
                              float* __restrict__ D) {         // [12800, 512]
    const int M   = B_ * S_;        // 12800
    const int NT2 = H_ / 32;        // 16 super-tiles in N
    int wid  = blockIdx.x * (blockDim.x >> 5) + (threadIdx.x >> 5);
    int lane = threadIdx.x & 31;
    int total = (M / 32) * NT2;     // 6400
    if (wid >= total) return;

    int mt = wid / NT2, nt = wid % NT2;
    int m0 = mt * 32, n0 = nt * 32;
    int lh = lane >> 4;             // 0: lanes 0-15, 1: lanes 16-31
    int ll = lane & 15;

    v8f acc00, acc01, acc10, acc11;
    float bv0 = bias[n0 + ll];
    float bv1 = bias[n0 + 16 + ll];
#pragma unroll
    for (int v = 0; v < 8; ++v) {
        acc00[v] = bv0; acc01[v] = bv1;
        acc10[v] = bv0; acc11[v] = bv1;
    }

    for (int k = 0; k < H_; k += 4) {
        // A frags: lane holds A[m][k + 2*lh + {0,1}] (consecutive K pair)
        v2f a0 = *reinterpret_cast<const v2f*>(A + (size_t)(m0 + ll) * H_ + k + 2 * lh);
        v2f a1 = *reinterpret_cast<const v2f*>(A + (size_t)(m0 + 16 + ll) * H_ + k + 2 * lh);
        // B frags: lane holds B[k + 2*lh + {0,1}][n]
        int kr = k + 2 * lh;
        const float* w0 = Wb + (size_t)kr * H_;
        const float* w1 = Wb + (size_t)(kr + 1) * H_;
        v2f b0; b0[0] = w0[n0 + ll];      b0[1] = w1[n0 + ll];
        v2f b1; b1[0] = w0[n0 + 16 + ll]; b1[1] = w1[n0 + 16 + ll];
        acc00 = __builtin_amdgcn_wmma_f32_16x16x4_f32(false, a0, false, b0, (short)0, acc00, false, false);
        acc01 = __builtin_amdgcn_wmma_f32_16x16x4_f32(false, a0, false, b1, (short)0, acc01, false, false);
        acc10 = __builtin_amdgcn_wmma_f32_16x16x4_f32(false, a1, false, b0, (short)0, acc10, false, false);
        acc11 = __builtin_amdgcn_wmma_f32_16x16x4_f32(false, a1, false, b1, (short)0, acc11, false, false);
    }
#pragma unroll
    for (int v = 0; v < 8; ++v) {
        int r0 = m0 + v + 8 * lh;           // C/D layout: M = vgpr + 8*(lane>=16)
        int r1 = r0 + 16;
        D[(size_t)r0 * H_ + n0 + ll]      = acc00[v];
        D[(size_t)r0 * H_ + n0 + 16 + ll] = acc01[v];
        D[(size_t)r1 * H_ + n0 + ll]      = acc10[v];
        D[(size_t)r1 * H_ + n0 + 16 + ll] = acc11[v];
    }
}

// ---------------------------------------------------------------------------
// Fused per-step cell: embedding lookup, reduce, LSTM gates, attention
// (energy -> masked softmax -> context), and tanh(cat) projection.
// One block per batch row; all phases only touch row b (barriers between).
// ---------------------------------------------------------------------------
__global__ void step_cell(const int* __restrict__ trg_seq, int t,
                          const float* __restrict__ emb_table,
                          const float* __restrict__ W_red, const float* __restrict__ b_red,
                          const float* __restrict__ W_ih,  const float* __restrict__ W_hh,
                          const float* __restrict__ b_ih,  const float* __restrict__ b_hh,
                          const float* __restrict__ W_cat, const float* __restrict__ b_cat,
                          const float* __restrict__ mem,   const int* __restrict__ mask,
                          float* __restrict__ h, float* __restrict__ c,
                          float* __restrict__ ctx,
                          float* __restrict__ energy_out,
                          float* __restrict__ logit_in) {
    int b   = blockIdx.x;
    int tid = threadIdx.x;
    __shared__ float semb[E_];
    __shared__ float sctx[H_];
    __shared__ float sh[H_];
    __shared__ float sx[E_];
    __shared__ float sg[4 * H_];
    __shared__ float se[S_];
    __shared__ float red[256];

    // ---- load state ----
    int y = trg_seq[b * T_ + t];
    for (int k = tid; k < E_; k += blockDim.x) semb[k] = emb_table[(size_t)y * E_ + k];
    for (int k = tid; k < H_; k += blockDim.x) {
        sctx[k] = ctx[b * H_ + k];
        sh[k]   = h[b * H_ + k];
    }
    __syncthreads();

    // ---- x = concat(emb, ctx) @ W_red + b_red ----
    for (int j = tid; j < E_; j += blockDim.x) {
        float acc = b_red[j];
        for (int k = 0; k < E_; ++k)  acc += semb[k] * W_red[(size_t)k * E_ + j];
        for (int k = 0; k < H_; ++k)  acc += sctx[k] * W_red[(size_t)(E_ + k) * E_ + j];
        sx[j] = acc;
    }
    __syncthreads();

    // ---- gates = x @ W_ih + h @ W_hh + b_ih + b_hh ----
    for (int j = tid; j < 4 * H_; j += blockDim.x) {
        float acc = b_ih[j] + b_hh[j];
        for (int k = 0; k < E_; ++k) acc += sx[k] * W_ih[(size_t)k * (4 * H_) + j];
        for (int k = 0; k < H_; ++k) acc += sh[k] * W_hh[(size_t)k * (4 * H_) + j];
        sg[j] = acc;
    }
    __syncthreads();

    // ---- LSTM cell (torch gate order i, f, g, o); sh <- h_new ----
    for (int j = tid; j < H_; j += blockDim.x) {
        float ig = sg[j];
        float fg = sg[H_ + j];
        float gg = sg[2 * H_ + j];
        float og = sg[3 * H_ + j];
        float cn = sigmoidf_(fg) * c[b * H_ + j] + sigmoidf_(ig) * tanhf(gg);
        float hn = sigmoidf_(og) * tanhf(cn);
        c[b * H_ + j] = cn;
        h[b * H_ + j] = hn;
        sh[j] = hn;
    }
    __syncthreads();

    // ---- attention energies (masked) ----
    for (int s = tid; s < S_; s += blockDim.x) {
        float e;
        if (mask[b * S_ + s] == 0) {
            e = NEG_INF;
        } else {
            const float* mp = mem + ((size_t)b * S_ + s) * H_;
            float acc = 0.0f;
            for (int k = 0; k < H_; ++k) acc += sh[k] * mp[k];
            e = acc;
        }
        se[s] = e;
        energy_out[b * S_ + s] = e;
    }
    __syncthreads();

    // ---- softmax over S ----
    float lm = NEG_INF;
    for (int s = tid; s < S_; s += blockDim.x) lm = fmaxf(lm, se[s]);
    red[tid] = lm;
    __syncthreads();
    for (int off = 128; off > 0; off >>= 1) {
        if (tid < off) red[tid] = fmaxf(red[tid], red[tid + off]);
        __syncthreads();
    }
    float mx = red[0];
    __syncthreads();
    float ls = 0.0f;
    for (int s = tid; s < S_; s += blockDim.x) {
        float ev = expf(se[s] - mx);
        se[s] = ev;
        ls += ev;
    }
    red[tid] = ls;
    __syncthreads();
    for (int off = 128; off > 0; off >>= 1) {
        if (tid < off) red[tid] += red[tid + off];
        __syncthreads();
    }
    float inv = 1.0f / red[0];
    __syncthreads();
    for (int s = tid; s < S_; s += blockDim.x) se[s] *= inv;
    __syncthreads();

    // ---- ctx = attn @ memories (lane-contiguous over H) ; sctx <- ctx_new ----
    for (int k = tid; k < H_; k += blockDim.x) {
        float acc = 0.0f;
        for (int s = 0; s < S_; ++s)
            acc += se[s] * mem[((size_t)b * S_ + s) * H_ + k];
        ctx[b * H_ + k] = acc;
        sctx[k] = acc;
    }
    __syncthreads();

    // ---- logit_in = tanh(concat(h_new, ctx_new) @ W_cat + b_cat) ----
    for (int j = tid; j < H_; j += blockDim.x) {
        float acc = b_cat[j];
        for (int k = 0; k < H_; ++k) acc += sh[k]   * W_cat[(size_t)k * H_ + j];
        for (int k = 0; k < H_; ++k) acc += sctx[k] * W_cat[(size_t)(H_ + k) * H_ + j];
        logit_in[b * H_ + j] = tanhf(acc);
    }
}

// ---------------------------------------------------------------------------
// Vocab projection: d_out[b, t, 0:V] = logit_in[32,512] @ W_log[512,30000] + b_log
// fp32 WMMA 16x16x4; each wave computes BOTH M-tiles (M=32) for its N-tile,
// so every W_log fragment load feeds two WMMAs (halves dominant L2 traffic).
// ---------------------------------------------------------------------------
__global__ void logit_wmma(const float* __restrict__ A,      // [32, 512]
                           const float* __restrict__ Wl,     // [512, 30000]
                           const float* __restrict__ bl,     // [30000]
                           float* __restrict__ out, int t) { // [B, T, VE]
    const int NT = V_ / 16;                   // 1875
    int wid  = blockIdx.x * (blockDim.x >> 5) + (threadIdx.x >> 5);
    int lane = threadIdx.x & 31;
    if (wid >= NT) return;

    int n0 = wid * 16;
    int lh = lane >> 4;
    int ll = lane & 15;

    v8f acc0, acc1;
    float bv = bl[n0 + ll];
#pragma unroll
    for (int v = 0; v < 8; ++v) { acc0[v] = bv; acc1[v] = bv; }

    for (int k = 0; k < H_; k += 4) {
        v2f a0 = *reinterpret_cast<const v2f*>(A + (size_t)(ll)      * H_ + k + 2 * lh);
        v2f a1 = *reinterpret_cast<const v2f*>(A + (size_t)(16 + ll) * H_ + k + 2 * lh);
        int kr = k + 2 * lh;
        v2f b;
        b[0] = Wl[(size_t)kr * V_ + n0 + ll];
        b[1] = Wl[(size_t)(kr + 1) * V_ + n0 + ll];
        acc0 = __builtin_amdgcn_wmma_f32_16x16x4_f32(false, a0, false, b, (short)0, acc0, false, false);
        acc1 = __builtin_amdgcn_wmma_f32_16x16x4_f32(false, a1, false, b, (short)0, acc1, false, false);
    }
#pragma unroll
    for (int v = 0; v < 8; ++v) {
        int r0 = v + 8 * lh;                  // batch rows 0-15
        int r1 = r0 + 16;                     // batch rows 16-31
        out[((size_t)r0 * T_ + t) * VE_ + n0 + ll] = acc0[v];
        out[((size_t)r1 * T_ + t) * VE_ + n0 + ll] = acc1[v];
    }
}

// ---------------------------------------------------------------------------
// Pointer mechanism (scatter-max of masked energies into extended vocab)
// ---------------------------------------------------------------------------
__global__ void ptr_fill(float* __restrict__ table) {
    int i = blockIdx.x * blockDim.x + threadIdx.x;
    if (i < B_ * VE_) table[i] = NEG_INF;
}

__global__ void ptr_scatter(const int* __restrict__ ext_src,   // [B,S]
                            const float* __restrict__ energy,  // [B,S]
                            float* __restrict__ table) {       // [B,VE]
    int i = blockIdx.x * blockDim.x + threadIdx.x;
    if (i >= B_ * S_) return;
    int b = i / S_;
    int pos = ext_src[i];
    atomicMax(&table[(size_t)b * VE_ + pos], energy[i]);
}

// Reads the scatter table, applies the ==-INF -> 0 and ==0 -> -INF rules,
// and RESETS each slot to -INF so the next step starts from a clean table.
__global__ void ptr_combine(float* __restrict__ table,         // [B,VE]
                            float* __restrict__ out, int t) {  // [B,T,VE]
    int i = blockIdx.x * blockDim.x + threadIdx.x;
    if (i >= B_ * VE_) return;
    int b = i / VE_;
    int n = i % VE_;
    float* row = out + ((size_t)b * T_ + t) * VE_;
    float ext = (n < V_) ? row[n] : 0.0f;       // OOV slots of ext_logit are 0
    float tv  = table[i];
    table[i]  = NEG_INF;                        // reset for next step
    float add = (tv == NEG_INF) ? 0.0f : tv;
    float val = ext + add;
    row[n] = (val == 0.0f) ? NEG_INF : val;
}

// ---------------------------------------------------------------------------
extern "C" void kernel_launch(void* const* d_in, const int* in_sizes, int n_in,
                              void* d_out, int out_size, void* d_ws, size_t ws_size,
                              hipStream_t stream) {
    const int*   trg_seq  = (const int*)  d_in[0];
    const int*   ext_src  = (const int*)  d_in[1];
    const float* h0       = (const float*)d_in[2];
    const float* c0       = (const float*)d_in[3];
    const float* enc_out  = (const float*)d_in[4];
    const int*   enc_mask = (const int*)  d_in[5];
    const float* emb      = (const float*)d_in[6];
    const float* W_enc    = (const float*)d_in[7];
    const float* b_enc    = (const float*)d_in[8];
    const float* W_red    = (const float*)d_in[9];
    const float* b_red    = (const float*)d_in[10];
    const float* W_ih     = (const float*)d_in[11];
    const float* W_hh     = (const float*)d_in[12];
    const float* b_ih     = (const float*)d_in[13];
    const float* b_hh     = (const float*)d_in[14];
    const float* W_cat    = (const float*)d_in[15];
    const float* b_cat    = (const float*)d_in[16];
    const float* W_log    = (const float*)d_in[17];
    const float* b_log    = (const float*)d_in[18];
    float* out = (float*)d_out;

    // Workspace layout (floats): ~30.4 MB total
    float* ws = (float*)d_ws;
    size_t off = 0;
    float* mem      = ws + off; off += (size_t)B_ * S_ * H_;  // 6,553,600
    float* h        = ws + off; off += (size_t)B_ * H_;
    float* c        = ws + off; off += (size_t)B_ * H_;
    float* ctx      = ws + off; off += (size_t)B_ * H_;
    float* energy   = ws + off; off += (size_t)B_ * S_;
    float* logit_in = ws + off; off += (size_t)B_ * H_;
    float* table    = ws + off; off += (size_t)B_ * VE_;
    (void)off; (void)ws_size; (void)n_in; (void)in_sizes; (void)out_size;

    // init state + clean scatter table (combine re-cleans it every step)
    init_state<<<(B_ * H_ + 255) / 256, 256, 0, stream>>>(h0, c0, h, c, ctx);
    ptr_fill<<<(B_ * VE_ + 255) / 256, 256, 0, stream>>>(table);

    // encoder features: one wave per 32x32 super-tile, 8 waves/block
    {
        int waves = (B_ * S_ / 32) * (H_ / 32);   // 6400
        memories_wmma<<<(waves + 7) / 8, 256, 0, stream>>>(enc_out, W_enc, b_enc, mem);
    }

    for (int t = 0; t < T_; ++t) {
        step_cell<<<B_, 256, 0, stream>>>(trg_seq, t, emb, W_red, b_red,
                                          W_ih, W_hh, b_ih, b_hh, W_cat, b_cat,
                                          mem, enc_mask, h, c, ctx, energy, logit_in);

        int waves = V_ / 16;                      // 1875 N-tiles, 8 waves/block
        logit_wmma<<<(waves + 7) / 8, 256, 0, stream>>>(logit_in, W_log, b_log, out, t);

        ptr_scatter<<<(B_ * S_ + 255) / 256, 256, 0, stream>>>(ext_src, energy, table);
        ptr_combine<<<(B_ * VE_ + 255) / 256, 256, 0, stream>>>(table, out, t);
    }
}